// MARN_73065983640197
// MI455X (gfx1250) — compile-verified
//
#include <hip/hip_runtime.h>
#include <math.h>

// MARN cell for MI455X (gfx1250, wave32).
// Memory-bound: ~800MB of M traffic => ~35us floor at 23.3 TB/s.
// GEMMs (~6.5 GFLOP) done with V_WMMA_F32_16X16X4_F32 (exact f32).
// M staging into LDS uses GLOBAL_LOAD_ASYNC_TO_LDS_B128 (ASYNCcnt path).

typedef __attribute__((ext_vector_type(2))) float v2f;
typedef __attribute__((ext_vector_type(8))) float v8f;
typedef __attribute__((ext_vector_type(4))) int   v4i;

#define B_      4096
#define IN_DIM_ 128
#define UNITS_  256
#define NUM_K_  64
#define NUM_S_  256
#define ROWS_   (IN_DIM_ + UNITS_)   // 384

#define AS1 __attribute__((address_space(1)))
#define AS3 __attribute__((address_space(3)))

#if __has_builtin(__builtin_amdgcn_global_load_async_to_lds_b128) && \
    __has_builtin(__builtin_amdgcn_s_wait_asynccnt)
#define HAVE_ASYNC_LDS 1
#endif

__device__ __forceinline__ float sigf(float x) { return 1.0f / (1.0f + __expf(-x)); }

__device__ __forceinline__ v8f wmma4(v2f a, v2f b, v8f c) {
  // D = A(16x4) * B(4x16) + C, all f32. 8-arg form:
  // (neg_a, A, neg_b, B, c_mod, C, reuse_a, reuse_b)
  return __builtin_amdgcn_wmma_f32_16x16x4_f32(false, a, false, b, (short)0, c,
                                               false, false);
}

// ---------------------------------------------------------------------------
// Kernel 1: four gate GEMMs  (xh[4096x384] @ W{i,f,o,t}[384x256]) with fused
// sigmoid epilogue producing c = f*c_prev + i*t  and o.
// Block: 128 threads = 4 waves; each wave: 16 batch rows x 16 cols, 4 gates.
// ---------------------------------------------------------------------------
__global__ __launch_bounds__(128) void gates_gemm(
    const float* __restrict__ X, const float* __restrict__ h_prev,
    const float* __restrict__ c_prev,
    const float* __restrict__ Wi, const float* __restrict__ Wf,
    const float* __restrict__ Wo, const float* __restrict__ Wt,
    const float* __restrict__ pbi, const float* __restrict__ pbf,
    const float* __restrict__ pbo, const float* __restrict__ pbt,
    float* __restrict__ out_c, float* __restrict__ ws_o)
{
  const int wave  = threadIdx.x >> 5;
  const int lane  = threadIdx.x & 31;
  const int row0  = blockIdx.x * 64 + wave * 16;
  const int col0  = blockIdx.y * 16;
  const int mA    = lane & 15;          // A row / C column index
  const int khalf = lane >> 4;          // which K pair this lane holds
  const int n     = col0 + mA;          // output column

  v8f acc_i = {}; v8f acc_f = {}; v8f acc_o = {}; v8f acc_t = {};

  const float* xrow = X      + (size_t)(row0 + mA) * IN_DIM_;
  const float* hrow = h_prev + (size_t)(row0 + mA) * UNITS_;
  const float* pWi = Wi + n; const float* pWf = Wf + n;
  const float* pWo = Wo + n; const float* pWt = Wt + n;

  #pragma unroll 4
  for (int k0 = 0; k0 < ROWS_; k0 += 4) {
    const int kk = k0 + khalf * 2;      // even; never straddles the 128 split
    v2f a = (kk < IN_DIM_) ? *(const v2f*)(xrow + kk)
                           : *(const v2f*)(hrow + (kk - IN_DIM_));
    const int w0 = kk * UNITS_, w1 = w0 + UNITS_;
    v2f fb_i = { pWi[w0], pWi[w1] };
    v2f fb_f = { pWf[w0], pWf[w1] };
    v2f fb_o = { pWo[w0], pWo[w1] };
    v2f fb_t = { pWt[w0], pWt[w1] };
    acc_i = wmma4(a, fb_i, acc_i);      // A fragment reused across 4 WMMAs
    acc_f = wmma4(a, fb_f, acc_f);
    acc_o = wmma4(a, fb_o, acc_o);
    acc_t = wmma4(a, fb_t, acc_t);
  }

  const float Bi = pbi[n], Bf = pbf[n], Bo = pbo[n], Bt = pbt[n];
  #pragma unroll
  for (int v = 0; v < 8; ++v) {
    const int row = row0 + v + khalf * 8;        // C/D layout: M = v + 8*(lane/16)
    const float iv = sigf(acc_i[v] + Bi);
    const float fv = sigf(acc_f[v] + Bf);
    const float ov = sigf(acc_o[v] + Bo);
    const float tv = sigf(acc_t[v] + Bt);
    const size_t idx = (size_t)row * UNITS_ + n;
    out_c[idx] = fv * c_prev[idx] + iv * tv;
    ws_o[idx]  = ov;
  }
}

// ---------------------------------------------------------------------------
// Kernel 2: attention over M.  One block (256 thr, 8 waves) per batch row.
// Stages M[b] (64KB) in LDS once (async global->LDS, ASYNCcnt) so that
// cos + softmax + r share a single HBM read.
// ---------------------------------------------------------------------------
__global__ __launch_bounds__(256) void attn_kernel(
    const float* __restrict__ M_prev, const float* __restrict__ k_prev,
    float* __restrict__ ws_alpha, float* __restrict__ ws_r)
{
  extern __shared__ float sm[];
  float* Ml     = sm;                       // 64*256
  float* kpL    = sm + NUM_K_ * NUM_S_;     // 256
  float* cosL   = kpL + NUM_S_;             // 64
  float* alphaL = cosL + NUM_K_;            // 64

  const int t = threadIdx.x;
  const int b = blockIdx.x;
  const float* Mb = M_prev + (size_t)b * NUM_K_ * NUM_S_;

  const float4* Ms4 = (const float4*)Mb;
  float4* Ml4 = (float4*)Ml;
#if defined(HAVE_ASYNC_LDS)
  // 256 threads x 16 x 16B async copies = 64KB, VGPR-free, ASYNCcnt-tracked.
  #pragma unroll
  for (int j = 0; j < 16; ++j)
    __builtin_amdgcn_global_load_async_to_lds_b128(
        (AS1 v4i*)(Ms4 + t + 256 * j), (AS3 v4i*)(Ml4 + t + 256 * j), 0, 0);
  kpL[t] = k_prev[(size_t)b * NUM_S_ + t];   // overlaps with async copies
  __builtin_amdgcn_s_wait_asynccnt(0);
  __syncthreads();
#else
  #pragma unroll
  for (int j = 0; j < 16; ++j) Ml4[t + 256 * j] = Ms4[t + 256 * j];
  kpL[t] = k_prev[(size_t)b * NUM_S_ + t];
  __syncthreads();
#endif

  const int lane = t & 31, wave = t >> 5;

  // ||k||^2 (each wave computes it redundantly via shuffles)
  float ssk = 0.f;
  #pragma unroll
  for (int j = 0; j < 8; ++j) { float v = kpL[lane + 32 * j]; ssk += v * v; }
  #pragma unroll
  for (int off = 16; off; off >>= 1) ssk += __shfl_xor(ssk, off, 32);
  const float inv_nk = rsqrtf(fmaxf(ssk, 1e-12f));

  // 8 memory rows per wave: normalized dot products
  for (int kr = wave * 8; kr < wave * 8 + 8; ++kr) {
    float d = 0.f, ss = 0.f;
    const float* Mr = Ml + kr * NUM_S_;
    #pragma unroll
    for (int j = 0; j < 8; ++j) {
      float m = Mr[lane + 32 * j];
      d  += m * kpL[lane + 32 * j];
      ss += m * m;
    }
    #pragma unroll
    for (int off = 16; off; off >>= 1) {
      d  += __shfl_xor(d, off, 32);
      ss += __shfl_xor(ss, off, 32);
    }
    if (lane == 0) cosL[kr] = d * rsqrtf(fmaxf(ss, 1e-12f)) * inv_nk;
  }
  __syncthreads();

  // softmax over 64 values of -cos (redundant per thread; tiny)
  float mx = -INFINITY;
  for (int k = 0; k < NUM_K_; ++k) mx = fmaxf(mx, -cosL[k]);
  float sum = 0.f;
  for (int k = 0; k < NUM_K_; ++k) sum += __expf(-cosL[k] - mx);
  const float inv = 1.f / sum;
  if (t < NUM_K_) {
    float al = __expf(-cosL[t] - mx) * inv;
    alphaL[t] = al;
    ws_alpha[(size_t)b * NUM_K_ + t] = al;
  }
  __syncthreads();

  // r[s] = sum_k alpha_k * M[k][s]   (s = t; conflict-free LDS column walk)
  float r = 0.f;
  for (int k = 0; k < NUM_K_; ++k) r += alphaL[k] * Ml[k * NUM_S_ + t];
  ws_r[(size_t)b * NUM_S_ + t] = r;
}

// ---------------------------------------------------------------------------
// Kernel 3/5: segmented 3x [4096x256]@[256x256] WMMA GEMM -> P[b][768].
// blockIdx.y in [0,48): 16 column-tiles per segment; segment picks A & W.
// ---------------------------------------------------------------------------
__global__ __launch_bounds__(128) void seg_gemm(
    const float* __restrict__ in0, const float* __restrict__ in1,
    const float* __restrict__ in2,
    const float* __restrict__ w0, const float* __restrict__ w1,
    const float* __restrict__ w2,
    float* __restrict__ outP)
{
  const int wave  = threadIdx.x >> 5;
  const int lane  = threadIdx.x & 31;
  const int row0  = blockIdx.x * 64 + wave * 16;
  const int colG  = blockIdx.y * 16 + (lane & 15);   // 0..767
  const int seg   = blockIdx.y >> 4;                 // uniform per block
  const int n     = colG - seg * UNITS_;
  const int khalf = lane >> 4;

  const float* Ain = (seg == 0) ? in0 : (seg == 1) ? in1 : in2;
  const float* W   = (seg == 0) ? w0  : (seg == 1) ? w1  : w2;
  const float* arow = Ain + (size_t)(row0 + (lane & 15)) * UNITS_;
  const float* pw = W + n;

  v8f acc = {};
  #pragma unroll 8
  for (int k0 = 0; k0 < UNITS_; k0 += 4) {
    const int kk = k0 + khalf * 2;
    v2f a = *(const v2f*)(arow + kk);
    v2f fb = { pw[kk * UNITS_], pw[kk * UNITS_ + UNITS_] };
    acc = wmma4(a, fb, acc);
  }
  #pragma unroll
  for (int v = 0; v < 8; ++v) {
    const int row = row0 + v + khalf * 8;
    outP[(size_t)row * 768 + colG] = acc[v];
  }
}

// ---------------------------------------------------------------------------
// Kernel 4: rc1 = sig(r@Wr + c@Wc);  h = o * tanh(c + rc1*(r@Wh))
// ---------------------------------------------------------------------------
__global__ __launch_bounds__(256) void ew_h_kernel(
    const float* __restrict__ P, const float* __restrict__ cbuf,
    const float* __restrict__ obuf, float* __restrict__ hout)
{
  const int idx = blockIdx.x * 256 + threadIdx.x;   // over B*256
  const int b = idx >> 8, n = idx & 255;
  const size_t pb = (size_t)b * 768;
  const float rc1 = sigf(P[pb + n] + P[pb + 256 + n]);
  hout[idx] = obuf[idx] * tanhf(cbuf[idx] + rc1 * P[pb + 512 + n]);
}

// Kernel 6: k_curr = tanh(h@Wk + bk)
__global__ __launch_bounds__(256) void ew_k_kernel(
    const float* __restrict__ Q, const float* __restrict__ bk,
    float* __restrict__ kout)
{
  const int idx = blockIdx.x * 256 + threadIdx.x;
  const int b = idx >> 8, n = idx & 255;
  kout[idx] = tanhf(Q[(size_t)b * 768 + n] + bk[n]);
}

// ---------------------------------------------------------------------------
// Kernel 7: M update, float4-vectorized stream (dominant HBM traffic).
// e,a computed once per thread; alpha broadcast through LDS.
// ---------------------------------------------------------------------------
__global__ __launch_bounds__(256) void m_update_kernel(
    const float* __restrict__ M_prev, const float* __restrict__ Q,
    const float* __restrict__ be, const float* __restrict__ ba,
    const float* __restrict__ ws_alpha, float* __restrict__ M_out)
{
  __shared__ float alphaL[NUM_K_];
  const int t = threadIdx.x;
  const int b = blockIdx.x;
  if (t < NUM_K_) alphaL[t] = ws_alpha[(size_t)b * NUM_K_ + t];
  __syncthreads();

  const int s4    = (t & 63) * 4;       // this thread's 4 s-columns
  const int kbase = t >> 6;             // 0..3 -> covers k = kbase + 4*j

  const float* qe = Q + (size_t)b * 768 + 256 + s4;
  const float* qa = Q + (size_t)b * 768 + 512 + s4;
  float4 e4, a4;
  e4.x = sigf(qe[0] + be[s4 + 0]); e4.y = sigf(qe[1] + be[s4 + 1]);
  e4.z = sigf(qe[2] + be[s4 + 2]); e4.w = sigf(qe[3] + be[s4 + 3]);
  a4.x = tanhf(qa[0] + ba[s4 + 0]); a4.y = tanhf(qa[1] + ba[s4 + 1]);
  a4.z = tanhf(qa[2] + ba[s4 + 2]); a4.w = tanhf(qa[3] + ba[s4 + 3]);

  const float4* Mp = (const float4*)(M_prev + (size_t)b * NUM_K_ * NUM_S_);
  float4*       Mo = (float4*)(M_out + (size_t)b * NUM_K_ * NUM_S_);

  #pragma unroll
  for (int j = 0; j < 16; ++j) {
    const int k = kbase + 4 * j;
    const float al = alphaL[k];
    const int idx4 = k * 64 + (t & 63);
    if (j < 15) __builtin_prefetch((const void*)(Mp + idx4 + 256), 0, 1);
    float4 m = Mp[idx4];
    float4 o;
    o.x = m.x * (1.f - al * e4.x) + al * a4.x;
    o.y = m.y * (1.f - al * e4.y) + al * a4.y;
    o.z = m.z * (1.f - al * e4.z) + al * a4.z;
    o.w = m.w * (1.f - al * e4.w) + al * a4.w;
    Mo[idx4] = o;
  }
}

// ---------------------------------------------------------------------------
extern "C" void kernel_launch(void* const* d_in, const int* in_sizes, int n_in,
                              void* d_out, int out_size, void* d_ws, size_t ws_size,
                              hipStream_t stream) {
  (void)in_sizes; (void)n_in; (void)out_size; (void)ws_size;
  const float* X      = (const float*)d_in[0];
  const float* h_prev = (const float*)d_in[1];
  const float* c_prev = (const float*)d_in[2];
  const float* M_prev = (const float*)d_in[3];
  const float* k_prev = (const float*)d_in[4];
  const float* Wf = (const float*)d_in[5];
  const float* Wt = (const float*)d_in[6];
  const float* Wi = (const float*)d_in[7];
  const float* Wo = (const float*)d_in[8];
  const float* bf = (const float*)d_in[9];
  const float* bt = (const float*)d_in[10];
  const float* bi = (const float*)d_in[11];
  const float* bo = (const float*)d_in[12];
  const float* Wk = (const float*)d_in[13];
  const float* bk = (const float*)d_in[14];
  const float* Wr = (const float*)d_in[15];
  const float* Wc = (const float*)d_in[16];
  const float* Wh = (const float*)d_in[17];
  const float* We = (const float*)d_in[18];
  const float* be = (const float*)d_in[19];
  const float* Wa = (const float*)d_in[20];
  const float* ba = (const float*)d_in[21];

  float* out = (float*)d_out;
  const size_t HC = (size_t)B_ * UNITS_;               // 1048576
  const size_t BM = (size_t)B_ * NUM_K_ * NUM_S_;      // 67108864
  float* out_h = out;
  float* out_c = out + HC;
  float* out_M = out + 2 * HC;
  float* out_k = out + 2 * HC + BM;

  float* ws       = (float*)d_ws;
  float* ws_o     = ws;                                // B*256
  float* ws_alpha = ws_o + HC;                         // B*64
  float* ws_r     = ws_alpha + (size_t)B_ * NUM_K_;    // B*256
  float* ws_P     = ws_r + HC;                         // B*768 (reused as Q)

  const int dynLDS = (NUM_K_ * NUM_S_ + NUM_S_ + 2 * NUM_K_) * (int)sizeof(float);
  (void)hipFuncSetAttribute((const void*)attn_kernel,
                            hipFuncAttributeMaxDynamicSharedMemorySize, dynLDS);

  dim3 gGates(B_ / 64, UNITS_ / 16);
  gates_gemm<<<gGates, 128, 0, stream>>>(X, h_prev, c_prev, Wi, Wf, Wo, Wt,
                                         bi, bf, bo, bt, out_c, ws_o);

  attn_kernel<<<B_, 256, dynLDS, stream>>>(M_prev, k_prev, ws_alpha, ws_r);

  dim3 gSeg(B_ / 64, 48);
  seg_gemm<<<gSeg, 128, 0, stream>>>(ws_r, out_c, ws_r, Wr, Wc, Wh, ws_P);

  ew_h_kernel<<<(B_ * UNITS_) / 256, 256, 0, stream>>>(ws_P, out_c, ws_o, out_h);

  seg_gemm<<<gSeg, 128, 0, stream>>>(out_h, out_h, out_h, Wk, We, Wa, ws_P);

  ew_k_kernel<<<(B_ * UNITS_) / 256, 256, 0, stream>>>(ws_P, bk, out_k);

  m_update_kernel<<<B_, 256, 0, stream>>>(M_prev, ws_P, be, ba, ws_alpha, out_M);
}